// SSRTEngine_49675591746432
// MI455X (gfx1250) — compile-verified
//
#include <hip/hip_runtime.h>
#include <hip/hip_bf16.h>
#include <stdint.h>

// ---------------------------------------------------------------------------
// SSRT on MI455X (gfx1250, wave32):
//  - TDM (tensor_load_to_lds) stages all 4 depth slices (256KB) into LDS once
//    per workgroup; hot march loop gathers depth from LDS (ds_load_b32).
//  - V_WMMA_F32_16X16X4_F32 performs the batched 4x4 projective transforms
//    (16 rays per WMMA, exact f32).
//  - Per-step divide hoisted to loop-invariant reciprocals.
// ---------------------------------------------------------------------------

#define SN_ 4
#define H_  128
#define W_  128
#define SLICE_ (H_ * W_)

typedef __attribute__((ext_vector_type(2)))  float    v2f;
typedef __attribute__((ext_vector_type(8)))  float    v8f;
typedef __attribute__((ext_vector_type(4)))  uint32_t v4u;
typedef __attribute__((ext_vector_type(4)))  int32_t  v4i;
typedef __attribute__((ext_vector_type(8)))  int32_t  v8i;

struct F4 { float x, y, z, w; };

// Batched 4x4 transform via WMMA: out[ray] = M(4x4) * in[ray] (homogeneous).
// A (16x4): rows 0..3 = matrix rows, rows 4..15 = 0.
//   Layout: lanes 0-15 hold M=lane, VGPR0=K0, VGPR1=K1; lanes 16-31 K2,K3.
// B (4x16): column n = ray n's (x,y,z,w).
//   Layout (mirrors A): lanes 0-15 = {K0,K1} of col=lane; lanes 16-31 = {K2,K3}.
// D (16x16 f32): VGPR m, lane n (n<16) = component m of ray n.
__device__ __forceinline__ F4 wave_transform(float ix, float iy, float iz, float iw,
                                             const float* __restrict__ M) {
  const int lane = (int)(threadIdx.x & 31u);
  const int hi   = lane >> 4;       // which half of the wave this lane is in
  const int m    = lane & 15;
  const int k0   = hi ? 2 : 0;

  float ax = 0.f, ay = 0.f;
  if (m < 4) { ax = M[m * 4 + k0]; ay = M[m * 4 + k0 + 1]; }
  v2f A = {ax, ay};

  // B operand for rays 0..15 (source lane = m)
  float x0 = __shfl(ix, m, 32),      y0 = __shfl(iy, m, 32);
  float z0 = __shfl(iz, m, 32),      w0 = __shfl(iw, m, 32);
  v2f B1 = { hi ? z0 : x0, hi ? w0 : y0 };
  // B operand for rays 16..31 (source lane = 16+m)
  float x1 = __shfl(ix, 16 + m, 32), y1 = __shfl(iy, 16 + m, 32);
  float z1 = __shfl(iz, 16 + m, 32), w1 = __shfl(iw, 16 + m, 32);
  v2f B2 = { hi ? z1 : x1, hi ? w1 : y1 };

  v8f C = {};
  v8f D1 = __builtin_amdgcn_wmma_f32_16x16x4_f32(false, A, false, B1, (short)0, C, false, false);
  v8f D2 = __builtin_amdgcn_wmma_f32_16x16x4_f32(false, A, false, B2, (short)0, C, false, false);

  // Lanes 0-15: own lane of D1 holds components 0..3 (VGPRs 0..3).
  // Lanes 16-31 (ray 16+m): component mm lives in D2[mm] at lane m.
  float o0b = __shfl(D1[0], lane, 32), o0c = __shfl(D2[0], m, 32);
  float o1b = __shfl(D1[1], lane, 32), o1c = __shfl(D2[1], m, 32);
  float o2b = __shfl(D1[2], lane, 32), o2c = __shfl(D2[2], m, 32);
  float o3b = __shfl(D1[3], lane, 32), o3c = __shfl(D2[3], m, 32);

  F4 o;
  o.x = hi ? o0c : o0b;
  o.y = hi ? o1c : o1b;
  o.z = hi ? o2c : o2b;
  o.w = hi ? o3c : o3b;
  return o;
}

// One-thread 4x4 inverse (adjugate), matching jnp.linalg.inv(proj) in f32.
__global__ void invert4x4_kernel(const float* __restrict__ a, float* __restrict__ out) {
  if (threadIdx.x != 0 || blockIdx.x != 0) return;
  float m[16];
  for (int i = 0; i < 16; ++i) m[i] = a[i];
  float inv[16];
  inv[0]  =  m[5]*m[10]*m[15] - m[5]*m[11]*m[14] - m[9]*m[6]*m[15] + m[9]*m[7]*m[14] + m[13]*m[6]*m[11] - m[13]*m[7]*m[10];
  inv[4]  = -m[4]*m[10]*m[15] + m[4]*m[11]*m[14] + m[8]*m[6]*m[15] - m[8]*m[7]*m[14] - m[12]*m[6]*m[11] + m[12]*m[7]*m[10];
  inv[8]  =  m[4]*m[9]*m[15]  - m[4]*m[11]*m[13] - m[8]*m[5]*m[15] + m[8]*m[7]*m[13] + m[12]*m[5]*m[11] - m[12]*m[7]*m[9];
  inv[12] = -m[4]*m[9]*m[14]  + m[4]*m[10]*m[13] + m[8]*m[5]*m[14] - m[8]*m[6]*m[13] - m[12]*m[5]*m[10] + m[12]*m[6]*m[9];
  inv[1]  = -m[1]*m[10]*m[15] + m[1]*m[11]*m[14] + m[9]*m[2]*m[15] - m[9]*m[3]*m[14] - m[13]*m[2]*m[11] + m[13]*m[3]*m[10];
  inv[5]  =  m[0]*m[10]*m[15] - m[0]*m[11]*m[14] - m[8]*m[2]*m[15] + m[8]*m[3]*m[14] + m[12]*m[2]*m[11] - m[12]*m[3]*m[10];
  inv[9]  = -m[0]*m[9]*m[15]  + m[0]*m[11]*m[13] + m[8]*m[1]*m[15] - m[8]*m[3]*m[13] - m[12]*m[1]*m[11] + m[12]*m[3]*m[9];
  inv[13] =  m[0]*m[9]*m[14]  - m[0]*m[10]*m[13] - m[8]*m[1]*m[14] + m[8]*m[2]*m[13] + m[12]*m[1]*m[10] - m[12]*m[2]*m[9];
  inv[2]  =  m[1]*m[6]*m[15]  - m[1]*m[7]*m[14]  - m[5]*m[2]*m[15] + m[5]*m[3]*m[14] + m[13]*m[2]*m[7]  - m[13]*m[3]*m[6];
  inv[6]  = -m[0]*m[6]*m[15]  + m[0]*m[7]*m[14]  + m[4]*m[2]*m[15] - m[4]*m[3]*m[14] - m[12]*m[2]*m[7]  + m[12]*m[3]*m[6];
  inv[10] =  m[0]*m[5]*m[15]  - m[0]*m[7]*m[13]  - m[4]*m[1]*m[15] + m[4]*m[3]*m[13] + m[12]*m[1]*m[7]  - m[12]*m[3]*m[5];
  inv[14] = -m[0]*m[5]*m[14]  + m[0]*m[6]*m[13]  + m[4]*m[1]*m[14] - m[4]*m[2]*m[13] - m[12]*m[1]*m[6]  + m[12]*m[2]*m[5];
  inv[3]  = -m[1]*m[6]*m[11]  + m[1]*m[7]*m[10]  + m[5]*m[2]*m[11] - m[5]*m[3]*m[10] - m[9]*m[2]*m[7]   + m[9]*m[3]*m[6];
  inv[7]  =  m[0]*m[6]*m[11]  - m[0]*m[7]*m[10]  - m[4]*m[2]*m[11] + m[4]*m[3]*m[10] + m[8]*m[2]*m[7]   - m[8]*m[3]*m[6];
  inv[11] = -m[0]*m[5]*m[11]  + m[0]*m[7]*m[9]   + m[4]*m[1]*m[11] - m[4]*m[3]*m[9]  - m[8]*m[1]*m[7]   + m[8]*m[3]*m[5];
  inv[15] =  m[0]*m[5]*m[10]  - m[0]*m[6]*m[9]   - m[4]*m[1]*m[10] + m[4]*m[2]*m[9]  + m[8]*m[1]*m[6]   - m[8]*m[2]*m[5];
  float det = m[0]*inv[0] + m[1]*inv[4] + m[2]*inv[8] + m[3]*inv[12];
  float rdet = 1.0f / det;
  for (int i = 0; i < 16; ++i) out[i] = inv[i] * rdet;
}

__global__ void __launch_bounds__(512, 1)
ssrt_main(const float* __restrict__ depth, const float* __restrict__ normal,
          const int* __restrict__ indices, const float* __restrict__ proj,
          const int* __restrict__ xin, const int* __restrict__ yin,
          const float* __restrict__ din, const float* __restrict__ dstart,
          const float* __restrict__ unproj, float* __restrict__ out, int bn) {
  extern __shared__ float sdepth[];  // SN_*H_*W_ floats = 256KB

  // ---- Stage all 4 depth slices into LDS with the Tensor Data Mover ----
  if (threadIdx.x == 0) {
    for (int s = 0; s < SN_; ++s) {
      uint32_t lds_off = (uint32_t)(uintptr_t)(&sdepth[s * SLICE_]);
      uint64_t gaddr   = (uint64_t)(uintptr_t)(depth + s * SLICE_);
      v4u g0;
      g0[0] = 1u;                                   // count=1, user descriptor
      g0[1] = lds_off;                              // lds_addr (bytes)
      g0[2] = (uint32_t)(gaddr & 0xFFFFFFFFull);    // global_addr[31:0]
      g0[3] = (uint32_t)((gaddr >> 32) & 0x01FFFFFFull) | (2u << 30);  // addr hi | type=2
      v8i g1;
      g1[0] = (int32_t)(2u << 16);        // data_size=2 (4B), no mask/pad/iter
      g1[1] = (int32_t)(128u << 16);      // tensor_dim0 = 128
      g1[2] = (int32_t)(128u << 16);      // tensor_dim1 = 128
      g1[3] = (int32_t)(128u << 16);      // tile_dim0   = 128
      g1[4] = (int32_t)(128u);            // tile_dim1   = 128, tile_dim2 = 0
      g1[5] = (int32_t)(128u);            // tensor_dim0_stride = 128
      g1[6] = (int32_t)(16384u << 16);    // tensor_dim1_stride[15:0] = 16384
      g1[7] = 0;
      v4i g2 = {0, 0, 0, 0};
      v4i g3 = {0, 0, 0, 0};
      v8i g4 = {0, 0, 0, 0, 0, 0, 0, 0};  // extra group in 6-arg (clang-23) form
      __builtin_amdgcn_tensor_load_to_lds(g0, g1, g2, g3, g4, 0);
    }
  }
  __builtin_amdgcn_s_wait_tensorcnt(0);
  __syncthreads();

  int i0 = (int)(blockIdx.x * blockDim.x + threadIdx.x);
  int i  = i0 < bn ? i0 : bn - 1;  // keep EXEC full for WMMA; stores guarded below

  // ---- Per-ray setup (reference: uv, proj_pos, view_pos, d_proj) ----
  const float xi = (float)xin[i];
  const float yi = (float)yin[i];
  const float uvx = (xi + 0.5f) * (2.0f / W_) - 1.0f;
  const float uvy = 1.0f - (yi + 0.5f) * (2.0f / H_);      // (1-(y+.5)/h)*2-1
  const float ds  = dstart[i] * 2.0f - 1.0f;
  const float d0 = din[i * 3 + 0], d1 = din[i * 3 + 1], d2 = din[i * 3 + 2];

  // view_pos = unproj * (uvx,uvy,ds,1), perspective divide
  F4 v = wave_transform(uvx, uvy, ds, 1.0f, unproj);
  float rw = 1.0f / v.w;
  float vx = v.x * rw, vy = v.y * rw, vz = v.z * rw;

  // d_proj = proj*(view + d*1e-3) / w - proj_pos
  F4 p = wave_transform(vx + d0 * 0.001f, vy + d1 * 0.001f, vz + d2 * 0.001f, 1.0f, proj);
  rw = 1.0f / p.w;
  const float ddx = p.x * rw - uvx;
  const float ddy = p.y * rw - uvy;
  const float ddz = p.z * rw - ds;

  float px = (uvx + 1.0f) * 0.5f;
  float py = (uvy + 1.0f) * 0.5f;
  float pz = (ds + 1.0f) * 0.5f;
  const int sx = (ddx >= 0.0f) ? 1 : -1;
  const int sy = (ddy >= 0.0f) ? -1 : 1;
  const float inv_dx = 1.0f / ddx;   // loop-invariant reciprocals (hoisted divide)
  const float inv_dy = 1.0f / ddy;

  int cx = (int)xi, cy = (int)yi;
  {  // initial march step (reference does one _march_next before the loop)
    float r0 = ((float)(cx + sx) * (1.0f / W_) - px) * inv_dx;
    float r1 = ((1.0f - (float)(cy + sy) * (1.0f / H_)) - py) * inv_dy;
    bool take_y = r0 > r1;
    float t = fminf(r0, r1);
    px += ddx * t; py += ddy * t; pz += ddz * t;
    cx += take_y ? 0 : sx;
    cy += take_y ? sy : 0;
  }

  const int idx = indices[i];
  const float* __restrict__ sd = &sdepth[idx * SLICE_];
  const float* __restrict__ nrm_base = normal + (size_t)idx * 3 * SLICE_;

  float res0 = 0.f, res1 = 0.f, hitf = 0.f, dzv = 0.f;

  // Monotone DDA: exits pixel bounds in <= W_+H_+2 steps; cap is a safety net.
  for (int it = 0; it < 2048; ++it) {
    if ((unsigned)cx >= (unsigned)W_ || (unsigned)cy >= (unsigned)H_) break;
    const int pix = cy * W_ + cx;
    const float z = sd[pix];                 // LDS gather (ds_load_b32)
    if (pz >= z) {                           // candidate hit: rare normal fetch
      const float n0 = nrm_base[pix];
      const float n1 = nrm_base[pix + SLICE_];
      const float n2 = nrm_base[pix + 2 * SLICE_];
      if (d0 * n0 + d1 * n1 + d2 * n2 <= 0.0f) {
        res0 = (float)cx; res1 = (float)cy; hitf = 1.0f; dzv = pz - z;
        break;
      }
    }
    float r0 = ((float)(cx + sx) * (1.0f / W_) - px) * inv_dx;
    float r1 = ((1.0f - (float)(cy + sy) * (1.0f / H_)) - py) * inv_dy;
    bool take_y = r0 > r1;
    float t = fminf(r0, r1);
    px += ddx * t; py += ddy * t; pz += ddz * t;
    cx += take_y ? 0 : sx;
    cy += take_y ? sy : 0;
  }

  if (i0 < bn) {
    out[2 * i0 + 0]  = res0;   // results (BN,2)
    out[2 * i0 + 1]  = res1;
    out[2 * bn + i0] = hitf;   // mask (BN)
    out[3 * bn + i0] = dzv;    // dz (BN,1)
  }
}

extern "C" void kernel_launch(void* const* d_in, const int* in_sizes, int n_in,
                              void* d_out, int out_size, void* d_ws, size_t ws_size,
                              hipStream_t stream) {
  const float* depth   = (const float*)d_in[0];
  const float* normal  = (const float*)d_in[1];
  const int*   indices = (const int*)d_in[2];
  const float* proj    = (const float*)d_in[3];
  const int*   x       = (const int*)d_in[4];
  const int*   y       = (const int*)d_in[5];
  const float* d       = (const float*)d_in[6];
  const float* dstart  = (const float*)d_in[7];
  float*       unproj  = (float*)d_ws;
  const int    bn      = in_sizes[4];

  invert4x4_kernel<<<1, 1, 0, stream>>>(proj, unproj);

  const int    block = 512;
  const int    grid  = (bn + block - 1) / block;
  const size_t lds   = (size_t)SN_ * SLICE_ * sizeof(float);  // 256 KB / WG
  (void)hipFuncSetAttribute((const void*)ssrt_main,
                            hipFuncAttributeMaxDynamicSharedMemorySize, (int)lds);
  ssrt_main<<<grid, block, lds, stream>>>(depth, normal, indices, proj, x, y, d,
                                          dstart, unproj, (float*)d_out, bn);
}